// GCNLayer_53377853554877
// MI455X (gfx1250) — compile-verified
//
#include <hip/hip_runtime.h>
#include <hip/hip_bf16.h>
#include <stdint.h>

#define N_NODES 50000
#define N_EDGES 800000
#define IN_SIZE 256
#define OUT_SIZE 256
#define M_TILES (N_NODES / 16)   // 3125
#define O_TILES (OUT_SIZE / 16)  // 16

typedef __attribute__((ext_vector_type(2))) float v2f;
typedef __attribute__((ext_vector_type(8))) float v8f;

// ---------------- degree ----------------
__global__ void k_zero_deg(uint32_t* __restrict__ deg) {
    int i = blockIdx.x * blockDim.x + threadIdx.x;
    if (i < N_NODES) deg[i] = 0u;
}

__global__ void k_deg(const int* __restrict__ src, uint32_t* __restrict__ deg) {
    int i = blockIdx.x * blockDim.x + threadIdx.x;
    if (i < N_EDGES) atomicAdd(&deg[src[i]], 1u);
}

__global__ void k_dinv(const uint32_t* __restrict__ deg, float* __restrict__ dinv) {
    int i = blockIdx.x * blockDim.x + threadIdx.x;
    if (i < N_NODES) {
        float f = (float)deg[i];
        float r = rsqrtf(f);
        r = r * (1.5f - 0.5f * f * r * r);   // Newton step: match deg**-0.5
        dinv[i] = r;
    }
}

// ---------------- self-loop term: agg[i][:] = x[i][:] * dinv[i]^2 ----------------
__global__ void k_self(const float* __restrict__ x, const float* __restrict__ dinv,
                       float* __restrict__ agg) {
    int i = blockIdx.x * blockDim.x + threadIdx.x;   // float4 index
    if (i < N_NODES * (IN_SIZE / 4)) {
        int node = i >> 6;                           // 64 float4 per row
        float s = dinv[node];
        s = s * s;
        float4 v = ((const float4*)x)[i];
        v.x *= s; v.y *= s; v.z *= s; v.w *= s;
        ((float4*)agg)[i] = v;
    }
}

// ---------------- edge scatter: agg[src] += x[dst] * dinv[src]*dinv[dst] ----------------
// one wave32 per edge; lane covers 8 channels via two coalesced float4 loads
__global__ void k_edges(const float* __restrict__ x,
                        const int* __restrict__ src, const int* __restrict__ dst,
                        const float* __restrict__ dinv, float* __restrict__ agg) {
    int wid  = (blockIdx.x * blockDim.x + threadIdx.x) >> 5;  // edge id
    int lane = threadIdx.x & 31;
    if (wid >= N_EDGES) return;                               // wave-uniform
    int s = src[wid];
    int d = dst[wid];
    float coef = dinv[s] * dinv[d];
    const float4* xr = (const float4*)(x + (size_t)d * IN_SIZE);
    float*        ar = agg + (size_t)s * IN_SIZE;
    float4 v0 = xr[lane];        // channels lane*4 .. lane*4+3
    float4 v1 = xr[lane + 32];   // channels 128+lane*4 ..
    atomicAdd(&ar[lane * 4 + 0],       v0.x * coef);
    atomicAdd(&ar[lane * 4 + 1],       v0.y * coef);
    atomicAdd(&ar[lane * 4 + 2],       v0.z * coef);
    atomicAdd(&ar[lane * 4 + 3],       v0.w * coef);
    atomicAdd(&ar[128 + lane * 4 + 0], v1.x * coef);
    atomicAdd(&ar[128 + lane * 4 + 1], v1.y * coef);
    atomicAdd(&ar[128 + lane * 4 + 2], v1.z * coef);
    atomicAdd(&ar[128 + lane * 4 + 3], v1.w * coef);
}

// ---------------- GEMM: out = relu(agg @ W^T + b) via V_WMMA_F32_16X16X4_F32 ----------------
// One wave per 16x16 output tile.
// A (16x4 f32) layout: VGPR v, lane l -> A[m0 + l%16][k0 + v + 2*(l/16)]
// B (4x16 f32) layout: VGPR v, lane l -> B[k0 + v + 2*(l/16)][o0 + l%16] = W[o0+l%16][k0+v+2*(l/16)]
// => per k-step both fragments are one 8-byte v2f load per lane.
__global__ void k_gemm(const float* __restrict__ agg, const float* __restrict__ Wm,
                       const float* __restrict__ bias, float* __restrict__ out) {
    int t = (blockIdx.x * blockDim.x + threadIdx.x) >> 5;     // tile id, wave-uniform
    if (t >= M_TILES * O_TILES) return;
    int o_tile = t & (O_TILES - 1);
    int m_tile = t >> 4;
    int m0 = m_tile * 16;
    int o0 = o_tile * 16;

    int lane = threadIdx.x & 31;
    int lm = lane & 15;
    int hi = lane >> 4;

    const float* Arow = agg + (size_t)(m0 + lm) * IN_SIZE + 2 * hi;
    const float* Brow = Wm  + (size_t)(o0 + lm) * IN_SIZE + 2 * hi;

    v8f c = {};
#pragma unroll 8
    for (int k0 = 0; k0 < IN_SIZE; k0 += 4) {
        v2f a = *(const v2f*)(Arow + k0);
        v2f bfr = *(const v2f*)(Brow + k0);
        c = __builtin_amdgcn_wmma_f32_16x16x4_f32(
                /*neg_a=*/false, a, /*neg_b=*/false, bfr,
                /*c_mod=*/(short)0, c, /*reuse_a=*/false, /*reuse_b=*/false);
    }

    float bv = bias[o0 + lm];
#pragma unroll
    for (int v = 0; v < 8; ++v) {
        float val = c[v] + bv;                    // D row = m0 + v + 8*hi, col = o0 + lm
        val = val > 0.0f ? val : 0.0f;
        out[(size_t)(m0 + v + 8 * hi) * OUT_SIZE + (o0 + lm)] = val;
    }
}

extern "C" void kernel_launch(void* const* d_in, const int* in_sizes, int n_in,
                              void* d_out, int out_size, void* d_ws, size_t ws_size,
                              hipStream_t stream) {
    const float* x  = (const float*)d_in[0];
    const int*   ei = (const int*)d_in[1];           // [2, E]: src then dst
    const float* Wm = (const float*)d_in[2];         // [out, in]
    const float* b  = (const float*)d_in[3];
    float* out = (float*)d_out;

    const int* src = ei;
    const int* dst = ei + N_EDGES;

    // workspace layout (hipMalloc alignment -> all offsets 8B-aligned)
    uint32_t* deg  = (uint32_t*)d_ws;                        // N u32
    float*    dinv = (float*)d_ws + N_NODES;                 // N f32
    float*    agg  = (float*)d_ws + 2 * (size_t)N_NODES;     // N*256 f32

    k_zero_deg<<<(N_NODES + 255) / 256, 256, 0, stream>>>(deg);
    k_deg<<<(N_EDGES + 255) / 256, 256, 0, stream>>>(src, deg);
    k_dinv<<<(N_NODES + 255) / 256, 256, 0, stream>>>(deg, dinv);
    k_self<<<(N_NODES * (IN_SIZE / 4) + 255) / 256, 256, 0, stream>>>(x, dinv, agg);
    k_edges<<<N_EDGES / 8, 256, 0, stream>>>(x, src, dst, dinv, agg);   // 8 waves/block
    k_gemm<<<(M_TILES * O_TILES) / 8, 256, 0, stream>>>(agg, Wm, b, out);
}